// CrossAttention_6150393168709
// MI455X (gfx1250) — compile-verified
//
#include <hip/hip_runtime.h>

// Shapes from the reference
#define BATCH 8
#define LQ   512
#define KK   512
#define DD   64
#define NN   64
#define MM   64
#define PP   64

typedef __attribute__((ext_vector_type(2))) float v2f;
typedef __attribute__((ext_vector_type(8))) float v8f;

// ---------------------------------------------------------------------------
// Kernel 1: vkc[b,k,n] = sum_p vk[b,k,p,n] * vexp[b,k,p]
// Streams 67MB of vk. One 64-thread block covers four (b,k) pairs; each lane
// owns 4 consecutive n (float4), so every p-iteration is a fully-coalesced
// 4x256B row read with 4x fewer VMEM instructions than scalar loads.
// vexp is staged once through LDS.
// ---------------------------------------------------------------------------
__global__ __launch_bounds__(64) void vkc_kernel(const float* __restrict__ vk,
                                                 const float* __restrict__ vexp,
                                                 float* __restrict__ vkc) {
  __shared__ float sE[4 * PP];
  const int tid  = threadIdx.x;
  const int pair = tid >> 4;                 // which of 4 (b,k) pairs
  const int nq   = (tid & 15) * 4;           // n quad
  const int bk0  = blockIdx.x * 4;

  for (int i = tid; i < 4 * PP; i += 64) sE[i] = vexp[(size_t)bk0 * PP + i];
  __syncthreads();

  const float* vkp = vk + (size_t)(bk0 + pair) * PP * NN + nq;
  float4 acc = {0.f, 0.f, 0.f, 0.f};
#pragma unroll 4
  for (int p = 0; p < PP; ++p) {
    const float  w = sE[pair * PP + p];
    const float4 v = *reinterpret_cast<const float4*>(vkp + (size_t)p * NN);
    acc.x = fmaf(w, v.x, acc.x);
    acc.y = fmaf(w, v.y, acc.y);
    acc.z = fmaf(w, v.z, acc.z);
    acc.w = fmaf(w, v.w, acc.w);
  }
  *reinterpret_cast<float4*>(vkc + (size_t)(bk0 + pair) * NN + nq) = acc;
}

// ---------------------------------------------------------------------------
// Kernel 2: per (b, 16-row L tile):
//   S = scale * q_tile(16x64) @ k^T(64x512)       [f32 WMMA 16x16x4]
//   W = exp(S - rowmax)  (unnormalized, rowsum kept)
//   tmp_tile = (W @ vkc(512x64)) / rowsum          [f32 WMMA 16x16x4]
// 128 threads = 4 waves. Scores live in LDS (16 x 516 f32).
// Dual accumulators break the WMMA RAW chain (2x shorter critical path).
// ---------------------------------------------------------------------------
#define SSTRIDE 516   // 512 + 4 pad -> bank spread (516 % 64 == 4)

__global__ __launch_bounds__(128) void attn_kernel(const float* __restrict__ q,
                                                   const float* __restrict__ kmat,
                                                   const float* __restrict__ vkc,
                                                   const float* __restrict__ scale_p,
                                                   float* __restrict__ tmp) {
  __shared__ float sS[16 * SSTRIDE];
  __shared__ float sRed[128];
  __shared__ float sRowMax[16];
  __shared__ float sRowSum[16];

  const int tid  = threadIdx.x;
  const int wave = tid >> 5;
  const int lane = tid & 31;
  const int half = lane >> 4;           // 0: lanes 0-15, 1: lanes 16-31
  const int lrow = lane & 15;           // M index (A/C) or N index (B)

  const int b  = blockIdx.x >> 5;       // L/16 == 32 tiles per batch
  const int lt = blockIdx.x & 31;
  const int l0 = lt * 16;

  const float scale = scale_p[0];

  // ---- Stage A: S = scale * q_tile @ k^T -------------------------------
  // A fragment (16x4 f32): vgpr0 = {K=0 | K=2}, vgpr1 = {K=1 | K=3}
  //   -> per lane a contiguous float2 at column (d0 + 2*half).
  const float* qrow = q + ((size_t)b * LQ + l0 + lrow) * DD + 2 * half;
  v2f afrag[16];
#pragma unroll
  for (int dd = 0; dd < 16; ++dd) {
    const float2 t = *reinterpret_cast<const float2*>(qrow + 4 * dd);
    afrag[dd].x = t.x; afrag[dd].y = t.y;
  }

  const float* kbase = kmat + (size_t)b * KK * DD;
  for (int ct = wave; ct < 32; ct += 4) {         // 8 column tiles per wave
    const int k0 = ct * 16;
    // B fragment (4x16): B[kk][n] = k[k0+n][d0+kk] -> contiguous float2 too.
    const float* krow = kbase + (size_t)(k0 + lrow) * DD + 2 * half;
    v8f acc0 = {0.f, 0.f, 0.f, 0.f, 0.f, 0.f, 0.f, 0.f};
    v8f acc1 = {0.f, 0.f, 0.f, 0.f, 0.f, 0.f, 0.f, 0.f};
#pragma unroll
    for (int dd = 0; dd < 16; dd += 2) {
      const float2 t0 = *reinterpret_cast<const float2*>(krow + 4 * dd);
      const float2 t1 = *reinterpret_cast<const float2*>(krow + 4 * dd + 4);
      v2f b0; b0.x = t0.x; b0.y = t0.y;
      v2f b1; b1.x = t1.x; b1.y = t1.y;
      acc0 = __builtin_amdgcn_wmma_f32_16x16x4_f32(false, afrag[dd],     false, b0,
                                                   (short)0, acc0, false, false);
      acc1 = __builtin_amdgcn_wmma_f32_16x16x4_f32(false, afrag[dd + 1], false, b1,
                                                   (short)0, acc1, false, false);
    }
    // C/D layout: vgpr r -> M = r + 8*half, N = lane&15
#pragma unroll
    for (int r = 0; r < 8; ++r) {
      const int m = r + 8 * half;
      sS[m * SSTRIDE + k0 + lrow] = (acc0[r] + acc1[r]) * scale;
    }
  }
  __syncthreads();

  // ---- Stage B: rowmax, exp, rowsum (normalization deferred) ----------
  {
    const int r = tid >> 3;   // row 0..15
    const int j = tid & 7;    // 8 threads per row, stride-8 cols (bank-friendly)
    float pmax = -3.4e38f;
    for (int i = 0; i < 64; ++i)
      pmax = fmaxf(pmax, sS[r * SSTRIDE + j + 8 * i]);
    sRed[tid] = pmax;
    __syncthreads();
    if (j == 0) {
      float mx = sRed[tid];
      for (int t = 1; t < 8; ++t) mx = fmaxf(mx, sRed[tid + t]);
      sRowMax[r] = mx;
    }
    __syncthreads();
    const float mx = sRowMax[r];
    float psum = 0.f;
    for (int i = 0; i < 64; ++i) {
      const int c = j + 8 * i;
      const float e = expf(sS[r * SSTRIDE + c] - mx);
      sS[r * SSTRIDE + c] = e;
      psum += e;
    }
    sRed[tid] = psum;
    __syncthreads();
    if (j == 0) {
      float s = sRed[tid];
      for (int t = 1; t < 8; ++t) s += sRed[tid + t];
      sRowSum[r] = s;
    }
  }
  __syncthreads();

  // ---- Stage C: tmp_tile = (Wexp @ vkc) / rowsum ----------------------
  {
    const int n0 = wave * 16;                    // each wave owns 16 N columns
    const float* vb = vkc + (size_t)b * KK * NN + n0 + lrow;
    v8f acc0 = {0.f, 0.f, 0.f, 0.f, 0.f, 0.f, 0.f, 0.f};
    v8f acc1 = {0.f, 0.f, 0.f, 0.f, 0.f, 0.f, 0.f, 0.f};
    for (int k0 = 0; k0 < KK; k0 += 8) {         // 2 independent WMMA chains
      const float2 ta =
          *reinterpret_cast<const float2*>(&sS[lrow * SSTRIDE + k0 + 2 * half]);
      const float2 tb =
          *reinterpret_cast<const float2*>(&sS[lrow * SSTRIDE + k0 + 4 + 2 * half]);
      v2f a0; a0.x = ta.x; a0.y = ta.y;
      v2f a1; a1.x = tb.x; a1.y = tb.y;
      v2f b0;
      b0.x = vb[(size_t)(k0 + 2 * half)     * NN];
      b0.y = vb[(size_t)(k0 + 2 * half + 1) * NN];
      v2f b1;
      b1.x = vb[(size_t)(k0 + 4 + 2 * half)     * NN];
      b1.y = vb[(size_t)(k0 + 4 + 2 * half + 1) * NN];
      acc0 = __builtin_amdgcn_wmma_f32_16x16x4_f32(false, a0, false, b0,
                                                   (short)0, acc0, false, false);
      acc1 = __builtin_amdgcn_wmma_f32_16x16x4_f32(false, a1, false, b1,
                                                   (short)0, acc1, false, false);
    }
#pragma unroll
    for (int r = 0; r < 8; ++r) {
      const int m = r + 8 * half;
      tmp[((size_t)b * LQ + l0 + m) * NN + n0 + lrow] =
          (acc0[r] + acc1[r]) / sRowSum[m];
    }
  }
}

// ---------------------------------------------------------------------------
// Kernel 3: attn_out[m] = sum_n vq[b,l,m,n]*tmp[b,l,n];
//           out = LayerNorm(q + attn_out)   (eps = 1e-3, keras default)
// Streams 67MB of vq; one 256-thread block per (b,l); 4 threads per row m,
// consecutive threads read consecutive 64B quarter-rows (fully coalesced).
// ---------------------------------------------------------------------------
__global__ __launch_bounds__(256) void out_kernel(const float* __restrict__ q,
                                                  const float* __restrict__ vq,
                                                  const float* __restrict__ tmp,
                                                  const float* __restrict__ gamma,
                                                  const float* __restrict__ beta,
                                                  float* __restrict__ out) {
  const int bl = blockIdx.x;                     // b*L + l
  __shared__ float sT[64];
  __shared__ float sPart[256];
  __shared__ float sRed[64];
  __shared__ float sRed2[64];
  __shared__ float sStat[2];

  const int tid = threadIdx.x;
  if (tid < 64) sT[tid] = tmp[(size_t)bl * NN + tid];
  __syncthreads();

  const int m    = tid >> 2;
  const int part = tid & 3;
  const float4* vqp = reinterpret_cast<const float4*>(
      vq + ((size_t)bl * MM + m) * NN + part * 16);
  float acc = 0.f;
#pragma unroll
  for (int i = 0; i < 4; ++i) {
    const float4 v = vqp[i];
    const int c = part * 16 + 4 * i;
    acc = fmaf(v.x, sT[c + 0], acc);
    acc = fmaf(v.y, sT[c + 1], acc);
    acc = fmaf(v.z, sT[c + 2], acc);
    acc = fmaf(v.w, sT[c + 3], acc);
  }
  sPart[tid] = acc;
  __syncthreads();

  float x = 0.f;
  if (part == 0) {
    x = sPart[tid] + sPart[tid + 1] + sPart[tid + 2] + sPart[tid + 3]
        + q[(size_t)bl * DD + m];                // residual add
    sRed[m]  = x;
    sRed2[m] = x * x;
  }
  __syncthreads();
  for (int s = 32; s >= 1; s >>= 1) {
    if (tid < s) { sRed[tid] += sRed[tid + s]; sRed2[tid] += sRed2[tid + s]; }
    __syncthreads();
  }
  if (tid == 0) {
    const float mu  = sRed[0]  * (1.f / 64.f);
    const float var = sRed2[0] * (1.f / 64.f) - mu * mu;
    sStat[0] = mu;
    sStat[1] = rsqrtf(var + 1e-3f);
  }
  __syncthreads();
  if (part == 0) {
    out[(size_t)bl * DD + m] = (x - sStat[0]) * sStat[1] * gamma[m] + beta[m];
  }
}

// ---------------------------------------------------------------------------
extern "C" void kernel_launch(void* const* d_in, const int* in_sizes, int n_in,
                              void* d_out, int out_size, void* d_ws, size_t ws_size,
                              hipStream_t stream) {
  const float* q     = (const float*)d_in[0];
  const float* kmat  = (const float*)d_in[1];
  const float* vq    = (const float*)d_in[2];
  const float* vk    = (const float*)d_in[3];
  const float* vexp  = (const float*)d_in[4];
  const float* scale = (const float*)d_in[5];
  const float* gamma = (const float*)d_in[6];
  const float* beta  = (const float*)d_in[7];
  float* out = (float*)d_out;

  // Workspace layout: vkc (B*K*N f32 = 1MB) | tmp (B*L*N f32 = 1MB)
  float* vkc = (float*)d_ws;
  float* tmp = vkc + (size_t)BATCH * KK * NN;

  vkc_kernel <<<BATCH * KK / 4,    64,  0, stream>>>(vk, vexp, vkc);
  attn_kernel<<<BATCH * (LQ / 16), 128, 0, stream>>>(q, kmat, vkc, scale, tmp);
  out_kernel <<<BATCH * LQ,        256, 0, stream>>>(q, vq, tmp, gamma, beta, out);
}